// ONNXCompatibleEdgeConv_64896955842761
// MI455X (gfx1250) — compile-verified
//
#include <hip/hip_runtime.h>

#define B_ 8
#define C_ 64
#define N_ 4096
#define OUT_ 64
#define K_ 20
#define EPS_ 1e-5f
#define SLOPE_ 0.2f
#define SSTRIDE 4100  // 4096 + pad; row base stays 16B-aligned (4100*4 % 16 == 0)

typedef __attribute__((ext_vector_type(2))) float v2f;
typedef __attribute__((ext_vector_type(4))) float v4f;
typedef __attribute__((ext_vector_type(8))) float v8f;

__device__ __forceinline__ v8f wmma_k4(v2f a, v2f b, v8f c) {
  // D(16x16,f32) = A(16x4,f32) x B(4x16,f32) + C
  return __builtin_amdgcn_wmma_f32_16x16x4_f32(false, a, false, b, (short)0, c,
                                               false, false);
}

__device__ __forceinline__ unsigned long long shflxor64(unsigned long long v, int m) {
  unsigned lo = (unsigned)v, hi = (unsigned)(v >> 32);
  lo = (unsigned)__shfl_xor((int)lo, m, 32);
  hi = (unsigned)__shfl_xor((int)hi, m, 32);
  return ((unsigned long long)hi << 32) | lo;
}

// ---- Kernel 0: x_t[b][n][c] = x[b][c][n]  (LDS transpose), fused xx ----
__global__ __launch_bounds__(256) void txx_kernel(const float* __restrict__ x,
                                                  float* __restrict__ xt,
                                                  float* __restrict__ xx) {
  __shared__ float tile[64][65];
  int b = blockIdx.y;
  int n0 = blockIdx.x * 64;
  int t = threadIdx.x;
  const float* xb = x + (size_t)b * C_ * N_;
  for (int i = t; i < 4096; i += 256) {  // coalesced reads along n
    int c = i >> 6, n = i & 63;
    tile[c][n] = xb[(size_t)c * N_ + n0 + n];
  }
  __syncthreads();
  float* xtb = xt + (size_t)b * N_ * C_ + (size_t)n0 * C_;
  for (int i = t; i < 4096; i += 256) {  // coalesced writes along c
    int n = i >> 6, c = i & 63;
    xtb[n * 64 + c] = tile[c][n];
  }
  if (t < 64) {  // xx[b][n0+t] = sum_c x^2
    float s = 0.f;
#pragma unroll
    for (int c = 0; c < 64; ++c) {
      float v = tile[c][t];
      s = fmaf(v, v, s);
    }
    xx[b * N_ + n0 + t] = s;
  }
}

// ------- Kernel 1: fused Gram (fp32 WMMA) + iterative top-K in LDS -------
// One block = batch b, 16 rows n0..n0+15. Score strip 16x4096 lives in LDS.
__global__ __launch_bounds__(256) void topk_kernel(const float* __restrict__ xt,
                                                   const float* __restrict__ xx,
                                                   int* __restrict__ idx_out) {
  extern __shared__ float sc[];  // 16 * SSTRIDE floats (262400 B dynamic LDS)
  const int b = blockIdx.y;
  const int n0 = blockIdx.x * 16;
  const int lane = threadIdx.x & 31;
  const int wave = threadIdx.x >> 5;  // 8 waves
  const int l15 = lane & 15;
  const int chOff = (lane >> 4) * 2;  // lanes 16-31 carry K+2,K+3
  const float* xtb = xt + (size_t)b * N_ * C_;

  // A: 16 rows (points n0..n0+15) x 64 channels; one b64 per K-step.
  const v2f* ap = (const v2f*)(xtb + (size_t)(n0 + l15) * C_ + chOff);
  v2f a[16];
#pragma unroll
  for (int j = 0; j < 16; ++j) a[j] = ap[2 * j];

  const float* xxb = xx + b * N_;
  for (int m0 = wave * 16; m0 < N_; m0 += 128) {  // waves stride column tiles
    int colB = m0 + l15;
    const v2f* bp = (const v2f*)(xtb + (size_t)colB * C_ + chOff);

    // Stage all 16 B operands first: issues a deep load clause so WMMAs
    // drain with progressive s_wait_loadcnt instead of full stalls.
    v2f bv[16];
#pragma unroll
    for (int j = 0; j < 16; ++j) bv[j] = bp[2 * j];
    float xm = xxb[colB];  // hoisted above WMMA chain

    // Branch-free prefetch of next tile's row (wrapped address, always valid)
    int mpre = (m0 + 128) & (N_ - 1);
    __builtin_prefetch(xtb + (size_t)(mpre + l15) * C_, 0, 3);

    v8f d = {0.f, 0.f, 0.f, 0.f, 0.f, 0.f, 0.f, 0.f};
#pragma unroll
    for (int j = 0; j < 16; ++j) d = wmma_k4(a[j], bv[j], d);

    // score = 2*dot - xx[m]  (row-constant -xx[n] dropped: rank-invariant)
#pragma unroll
    for (int v = 0; v < 8; ++v) {
      int row = v + ((lane >> 4) << 3);
      sc[row * SSTRIDE + colB] = fmaf(2.0f, d[v], -xm);
    }
  }
  __syncthreads();

  // Top-K: each wave owns 2 rows; 20 wave-wide argmax passes per row.
  for (int rr = 0; rr < 2; ++rr) {
    int row = wave * 2 + rr;
    float* srow = sc + row * SSTRIDE;
    int* orow = idx_out + (((size_t)b << 12) + (n0 + row)) * K_;
    for (int it = 0; it < K_; ++it) {
      unsigned long long best = 0ull;
      const v4f* srow4 = (const v4f*)srow;  // ds_load_b128 scans
      for (int c4 = lane; c4 < N_ / 4; c4 += 32) {
        v4f v = srow4[c4];
        int cbase = c4 * 4;
#pragma unroll
        for (int s = 0; s < 4; ++s) {
          unsigned u = __float_as_uint(v[s]);
          u = (u & 0x80000000u) ? ~u : (u | 0x80000000u);  // order-preserving
          unsigned long long key =
              ((unsigned long long)u << 32) | (unsigned)(~(cbase + s));
          if (key > best) best = key;
        }
      }
#pragma unroll
      for (int s = 16; s > 0; s >>= 1) {
        unsigned long long o = shflxor64(best, s);
        if (o > best) best = o;
      }
      if (lane == 0) {
        int col = (int)(~(unsigned)(best & 0xffffffffull));
        orow[it] = col;
        srow[col] = -__builtin_huge_valf();  // exclude from later passes
      }
      __syncthreads();  // uniform (2*20 per wave): LDS write visibility
    }
  }
}

// --- Kernel 2: P' = inv*(W1+W2)@x + bias, Q' = inv*W1@x (WMMA, BN folded) ---
// Outputs point-major (B,N,64) so kernel 3's gathers are fully coalesced.
__global__ __launch_bounds__(256) void pq_kernel(
    const float* __restrict__ xt, const float* __restrict__ W,
    const float* __restrict__ gamma, const float* __restrict__ beta,
    const float* __restrict__ rmean, const float* __restrict__ rvar,
    float* __restrict__ Pp, float* __restrict__ Qp) {
  __shared__ float A1[64 * 66];  // stride 66: 8B-aligned rows, few conflicts
  __shared__ float A2[64 * 66];
  __shared__ float invs[64];
  __shared__ float bias[64];
  int t = threadIdx.x;
  if (t < 64) {
    float inv = gamma[t] * rsqrtf(rvar[t] + EPS_);
    invs[t] = inv;
    bias[t] = beta[t] - rmean[t] * inv;
  }
  __syncthreads();
  for (int i = t; i < 4096; i += 256) {
    int o = i >> 6, c = i & 63;
    float inv = invs[o];
    float w1 = W[o * 128 + c];
    float w2 = W[o * 128 + 64 + c];
    A1[o * 66 + c] = inv * (w1 + w2);
    A2[o * 66 + c] = inv * w1;
  }
  __syncthreads();

  int lane = t & 31, wave = t >> 5;
  int l15 = lane & 15;
  int chOff = (lane >> 4) * 2;
  int b = blockIdx.y;
  int n0 = blockIdx.x * 128 + wave * 16;
  const float* xtb = xt + (size_t)b * N_ * C_;
  int colB = n0 + l15;

  const v2f* bp = (const v2f*)(xtb + (size_t)colB * C_ + chOff);
  v2f bv[16];
#pragma unroll
  for (int j = 0; j < 16; ++j) bv[j] = bp[2 * j];

  float* Pb = Pp + (size_t)b * N_ * OUT_;
  float* Qb = Qp + (size_t)b * N_ * OUT_;
  for (int o0 = 0; o0 < 64; o0 += 16) {
    // Stage LDS A-tiles first, then run the WMMA chains.
    v2f a1[16], a2[16];
#pragma unroll
    for (int j = 0; j < 16; ++j) {
      int ar = (o0 + l15) * 66 + 4 * j + chOff;  // even -> 8B aligned
      a1[j] = *(const v2f*)(A1 + ar);
      a2[j] = *(const v2f*)(A2 + ar);
    }
    v8f dP = {0.f, 0.f, 0.f, 0.f, 0.f, 0.f, 0.f, 0.f};
    v8f dQ = {0.f, 0.f, 0.f, 0.f, 0.f, 0.f, 0.f, 0.f};
#pragma unroll
    for (int j = 0; j < 16; ++j) {
      dP = wmma_k4(a1[j], bv[j], dP);
      dQ = wmma_k4(a2[j], bv[j], dQ);
    }
    // Per lane: 8 consecutive output channels at fixed point -> b128 stores.
    int obase = o0 + ((lane >> 4) << 3);
    float* pdst = Pb + (size_t)colB * OUT_ + obase;
    float* qdst = Qb + (size_t)colB * OUT_ + obase;
    v4f p0 = {dP[0] + bias[obase + 0], dP[1] + bias[obase + 1],
              dP[2] + bias[obase + 2], dP[3] + bias[obase + 3]};
    v4f p1 = {dP[4] + bias[obase + 4], dP[5] + bias[obase + 5],
              dP[6] + bias[obase + 6], dP[7] + bias[obase + 7]};
    v4f q0 = {dQ[0], dQ[1], dQ[2], dQ[3]};
    v4f q1 = {dQ[4], dQ[5], dQ[6], dQ[7]};
    *(v4f*)(pdst) = p0;
    *(v4f*)(pdst + 4) = p1;
    *(v4f*)(qdst) = q0;
    *(v4f*)(qdst + 4) = q1;
  }
}

// ---- Kernel 3: gather neighbors, h = P' - Q'[idx], leaky, max over K ----
__global__ __launch_bounds__(256) void edge_out_kernel(
    const int* __restrict__ idxw, const float* __restrict__ Pp,
    const float* __restrict__ Qp, float* __restrict__ out) {
  __shared__ float tile[64][33];
  int b = blockIdx.y;
  int n0 = blockIdx.x * 32;
  int lane = threadIdx.x & 31, wave = threadIdx.x >> 5;
  const float* Pb = Pp + (size_t)b * N_ * OUT_;
  const float* Qb = Qp + (size_t)b * N_ * OUT_;
#pragma unroll
  for (int i = 0; i < 4; ++i) {
    int nl = wave * 4 + i;
    int n = n0 + nl;
    int iv = 0;
    if (lane < K_) iv = idxw[(((size_t)b << 12) + n) * K_ + lane];
    // lane owns channels (2*lane, 2*lane+1): single b64 per gather
    v2f p = *(const v2f*)(Pb + (size_t)n * OUT_ + 2 * lane);
    float m0 = -__builtin_huge_valf(), m1 = m0;
#pragma unroll
    for (int k = 0; k < K_; ++k) {
      int m = __shfl(iv, k, 32);
      v2f q = *(const v2f*)(Qb + (size_t)m * OUT_ + 2 * lane);
      float h0 = p.x - q.x;
      float h1 = p.y - q.y;
      h0 = (h0 >= 0.f) ? h0 : SLOPE_ * h0;
      h1 = (h1 >= 0.f) ? h1 : SLOPE_ * h1;
      m0 = fmaxf(m0, h0);
      m1 = fmaxf(m1, h1);
    }
    tile[2 * lane][nl] = m0;
    tile[2 * lane + 1][nl] = m1;
  }
  __syncthreads();
  float* ob = out + (size_t)b * OUT_ * N_ + n0;  // coalesced (b,o,n) writes
  for (int o = wave; o < 64; o += 8) {
    ob[(size_t)o * N_ + lane] = tile[o][lane];
  }
}

extern "C" void kernel_launch(void* const* d_in, const int* in_sizes, int n_in,
                              void* d_out, int out_size, void* d_ws,
                              size_t ws_size, hipStream_t stream) {
  const float* x = (const float*)d_in[0];      // (8,64,4096)
  const float* W = (const float*)d_in[1];      // (64,128)
  const float* gamma = (const float*)d_in[2];  // (64)
  const float* beta = (const float*)d_in[3];
  const float* rmean = (const float*)d_in[4];
  const float* rvar = (const float*)d_in[5];
  float* out = (float*)d_out;  // (8,64,4096)

  // Workspace layout (~26.6 MB)
  float* xx = (float*)d_ws;                           // B*N floats
  int* idxw = (int*)(xx + B_ * N_);                   // B*N*K ints
  float* Pp = (float*)(idxw + (size_t)B_ * N_ * K_);  // B*N*64 floats
  float* Qp = Pp + (size_t)B_ * N_ * OUT_;            // B*N*64 floats
  float* xtw = Qp + (size_t)B_ * N_ * OUT_;           // B*N*64 floats

  txx_kernel<<<dim3(N_ / 64, B_), 256, 0, stream>>>(x, xtw, xx);
  pq_kernel<<<dim3(N_ / 128, B_), 256, 0, stream>>>(xtw, W, gamma, beta, rmean,
                                                    rvar, Pp, Qp);
  topk_kernel<<<dim3(N_ / 16, B_), 256, 16 * SSTRIDE * sizeof(float), stream>>>(
      xtw, xx, idxw);
  edge_out_kernel<<<dim3(N_ / 32, B_), 256, 0, stream>>>(idxw, Pp, Qp, out);
}